// FlatPoolIco_34411277976431
// MI455X (gfx1250) — compile-verified
//
#include <hip/hip_runtime.h>

// ---------------------------------------------------------------------------
// FlatPoolIco: icosahedral pad + 7-neighbor mean pool (stride 2) + pole clean.
// Memory-bound: ~315 MB traffic -> ~13.5us at 23.3 TB/s. Strategy: stage each
// chart-strip into LDS once via CDNA5 async global->LDS DMA, gather from LDS.
// ---------------------------------------------------------------------------

#define R_ROT 6
#define NCH   5
#define HH    128
#define WW    256
#define H2    64
#define W2    128
#define HT    16                 // output rows per block
#define LROWS (2 * HT + 1)       // 33 padded input rows staged in LDS
#define LPITCH 260               // floats per LDS row (16B aligned, bank-skewed)

#ifndef __has_builtin
#define __has_builtin(x) 0
#endif

#if __has_builtin(__builtin_amdgcn_global_load_async_to_lds_b128) && \
    __has_builtin(__builtin_amdgcn_s_wait_asynccnt)
#define HAVE_ASYNC 1
typedef int v4i_t __attribute__((ext_vector_type(4)));
#else
#define HAVE_ASYNC 0
#endif

__global__ __launch_bounds__(256) void flatpool_ico_kernel(
    const float* __restrict__ x, float* __restrict__ out) {
  __shared__ float tile[LROWS * LPITCH];
  __shared__ float haloL[LROWS];

  const int tid = threadIdx.x;
  int bid = blockIdx.x;
  const int hblk = bid & 3;           bid >>= 2;
  const int c    = bid % NCH;         bid /= NCH;
  const int Rr   = bid % R_ROT;
  const int b    = bid / R_ROT;

  const int h0     = hblk * HT;       // first output row of this strip
  const int phBase = 2 * h0;          // first padded row staged (LDS lr=0)

  const float* bBase  = x + (size_t)b * (R_ROT * NCH * HH * WW);
  const float* inBase = bBase + ((size_t)Rr * NCH + c) * (HH * WW);

  // ---- interior rows: padded row ph -> input row ph-1, cols 1..256 -> 0..255
  const int lrStart = (h0 == 0) ? 1 : 0;   // ph==0 is the stitched top pad row
  {
    const int cg = tid & 63;               // 64 groups of 4 floats = 256 cols
    for (int lr = lrStart + (tid >> 6); lr < LROWS; lr += 4) {
      const int ir = phBase + lr - 1;
      const float* src = inBase + (size_t)ir * WW + cg * 4;
      float* dst = &tile[lr * LPITCH + cg * 4];
#if HAVE_ASYNC
      __builtin_amdgcn_global_load_async_to_lds_b128(
          (v4i_t*)src, (v4i_t*)dst, 0, 0);
#else
      *(float4*)dst = *(const float4*)src;
#endif
    }
  }

  // ---- left pad column (pw==0): src = (R-1, c-1, 127, 128-ph); corner -> flat 0
  if (tid < LROWS) {
    const int ph = phBase + tid;
    float v;
    if (ph == 0) {
      v = bBase[0];  // reference quirk: unset _RIDX corner gathers flat index 0
    } else {
      const int srcW = 128 - ph;           // 0..127
      const float* p = bBase +
          (((size_t)((Rr + R_ROT - 1) % R_ROT)) * NCH + (c + NCH - 1) % NCH) *
              (size_t)(HH * WW) +
          (size_t)(HH - 1) * WW + srcW;
      v = (srcW == 0) ? 0.0f : *p;         // input pole (127,0) masked
    }
    haloL[tid] = v;
  }

  // ---- top pad row (ph==0), only for the h0==0 strip
  if (h0 == 0) {
    const int pw = tid + 1;                // 1..256
    float v;
    if (pw <= 128) {                       // (R, c-1, 127, 127+pw)
      v = bBase[((size_t)Rr * NCH + (c + NCH - 1) % NCH) * (size_t)(HH * WW) +
                (size_t)(HH - 1) * WW + (127 + pw)];
    } else {                               // (R+1, c-1, 256-pw, 255)
      const int srcH = 256 - pw;           // 127..0
      const float* p = bBase +
          (((size_t)((Rr + 1) % R_ROT)) * NCH + (c + NCH - 1) % NCH) *
              (size_t)(HH * WW) +
          (size_t)srcH * WW + (WW - 1);
      v = (srcH == 0) ? 0.0f : *p;         // input pole (0,255) masked
    }
    tile[pw - 1] = v;                      // row 0 of the tile
  }

#if HAVE_ASYNC
  __builtin_amdgcn_s_wait_asynccnt(0);     // our wave's async DMAs into LDS done
#endif
  __syncthreads();                         // all waves' fills visible

  // ---- in-chart pole pixels inside the staged interior: zero them
  if (tid == 0) {
    if (phBase == 0) tile[1 * LPITCH + 255] = 0.0f;   // src (0,255): ph=1,pw=256
    if (phBase + 32 == 128) tile[32 * LPITCH + 0] = 0.0f; // src (127,0): ph=128,pw=1
  }
  __syncthreads();

  // ---- 7-tap hex mean from LDS, 8 outputs per thread, coalesced stores
  float* obase = out +
      (((size_t)b * R_ROT + Rr) * NCH + c) * (size_t)(H2 * W2) +
      (size_t)h0 * W2;

  for (int i = 0; i < (HT * W2) / 256; ++i) {
    const int o  = i * 256 + tid;
    const int hl = o >> 7;                 // 0..15
    const int w  = o & (W2 - 1);           // 0..127
    const int lr0 = 2 * hl, lr1 = 2 * hl + 1, lr2 = 2 * hl + 2;
    const float* t0 = &tile[lr0 * LPITCH];
    const float* t1 = &tile[lr1 * LPITCH];
    const float* t2 = &tile[lr2 * LPITCH];
    // padded cols: pwL=2w, pwC=2w+1, pwR=2w+2 ; tile col = pw-1
    const int cC = 2 * w;                  // pwC-1
    const float vC0 = t0[cC];
    const float vC1 = t1[cC];
    const float vC2 = t2[cC];
    const float vR1 = t1[cC + 1];
    const float vR2 = t2[cC + 1];
    const float vL0 = (w == 0) ? haloL[lr0] : t0[cC - 1];
    const float vL1 = (w == 0) ? haloL[lr1] : t1[cC - 1];

    float res = (vC0 + vC1 + vC2 + vR1 + vR2 + vL0 + vL1) * (1.0f / 7.0f);

    const int h = h0 + hl;
    if ((h == H2 - 1 && w == 0) || (h == 0 && w == W2 - 1)) res = 0.0f; // out poles
    obase[(size_t)hl * W2 + w] = res;
  }
}

extern "C" void kernel_launch(void* const* d_in, const int* in_sizes, int n_in,
                              void* d_out, int out_size, void* d_ws, size_t ws_size,
                              hipStream_t stream) {
  const float* x = (const float*)d_in[0];
  float* out = (float*)d_out;
  const int B = in_sizes[0] / (R_ROT * NCH * HH * WW);   // 4*16 = 64
  const int nblocks = B * R_ROT * NCH * (H2 / HT);       // 7680
  flatpool_ico_kernel<<<nblocks, 256, 0, stream>>>(x, out);
  (void)n_in; (void)out_size; (void)d_ws; (void)ws_size;
}